// GCNConv_74594991997165
// MI455X (gfx1250) — compile-verified
//
#include <hip/hip_runtime.h>

typedef float v2f __attribute__((ext_vector_type(2)));
typedef float v8f __attribute__((ext_vector_type(8)));

// ---------------------------------------------------------------------------
// 1) deg[n] = 1.0 (self loop)
// ---------------------------------------------------------------------------
__global__ __launch_bounds__(256) void deg_init_kernel(float* __restrict__ deg, int N) {
    int n = blockIdx.x * 256 + threadIdx.x;
    if (n < N) deg[n] = 1.0f;
}

// ---------------------------------------------------------------------------
// 2) deg[col[e]] += 1 over all edges
// ---------------------------------------------------------------------------
__global__ __launch_bounds__(256) void deg_count_kernel(const int* __restrict__ cols,
                                                        float* __restrict__ deg, int E) {
    int e = blockIdx.x * 256 + threadIdx.x;
    if (e < E) atomicAdd(&deg[cols[e]], 1.0f);
}

// ---------------------------------------------------------------------------
// 3) dinv[n] = deg[n]^-1/2
// ---------------------------------------------------------------------------
__global__ __launch_bounds__(256) void dinv_kernel(const float* __restrict__ deg,
                                                   float* __restrict__ dinv, int N) {
    int n = blockIdx.x * 256 + threadIdx.x;
    if (n < N) dinv[n] = rsqrtf(deg[n]);
}

// ---------------------------------------------------------------------------
// 4) X_hat = X @ W^T via V_WMMA_F32_16X16X4_F32 (full fp32 matrix pipe).
//    One wave -> one 16x16 output tile. Block = 8 waves = one 16-row strip of
//    X across all 128 output columns (X rows are reused from WGP$).
//
//    Fragment layouts (wave32, ISA 7.12.2):
//      A 16x4 f32 (2 VGPR): lane m=lane%16; half=lane/16 selects K {0,1}/{2,3}
//      B 4x16  f32 (2 VGPR): lane n=lane%16; VGPR v: K = v + 2*half
//      C/D 16x16 f32 (8 VGPR): VGPR v: row = v + 8*half, col = lane%16
// ---------------------------------------------------------------------------
__global__ __launch_bounds__(256) void gemm_xwT_wmma_kernel(const float* __restrict__ X,
                                                            const float* __restrict__ W,
                                                            float* __restrict__ Xhat,
                                                            int N) {
    const int lane    = threadIdx.x & 31;
    const int wave    = threadIdx.x >> 5;        // 0..7 -> output col tile
    const int rowBase = blockIdx.x * 16;
    const int colBase = wave * 16;
    if (rowBase >= N) return;

    const int half = lane >> 4;                  // 0: lanes 0-15, 1: lanes 16-31
    const int l16  = lane & 15;

    // A: row (rowBase+l16) of X, K pair starting at 2*half
    const float* aptr = X + (size_t)(rowBase + l16) * 128 + (half << 1);
    // B[k][n] = W^T[k][n] = W[n][k]: row (colBase+l16) of W, K pair at 2*half
    const float* bptr = W + (size_t)(colBase + l16) * 128 + (half << 1);

    v8f acc = {0.f, 0.f, 0.f, 0.f, 0.f, 0.f, 0.f, 0.f};
#pragma unroll
    for (int k = 0; k < 128; k += 4) {
        v2f a = *(const v2f*)(aptr + k);
        v2f b = *(const v2f*)(bptr + k);
        acc = __builtin_amdgcn_wmma_f32_16x16x4_f32(
            /*neg_a=*/false, a, /*neg_b=*/false, b,
            /*c_mod=*/(short)0, acc, /*reuse_a=*/false, /*reuse_b=*/false);
    }

    float* cptr = Xhat + (size_t)rowBase * 128 + colBase;
#pragma unroll
    for (int v = 0; v < 8; ++v) {
        int m = v + (half << 3);
        cptr[(size_t)m * 128 + l16] = acc[v];    // 16 lanes -> 64B coalesced row
    }
}

// ---------------------------------------------------------------------------
// 5) out[n,f] = dinv[n]^2 * X_hat[n,f]   (self-loop term; also inits d_out)
// ---------------------------------------------------------------------------
__global__ __launch_bounds__(256) void out_init_kernel(const float* __restrict__ dinv,
                                                       const float* __restrict__ Xhat,
                                                       float* __restrict__ out, int total) {
    int i = blockIdx.x * 256 + threadIdx.x;
    if (i >= total) return;
    float d = dinv[i >> 7];
    out[i] = d * d * Xhat[i];
}

// ---------------------------------------------------------------------------
// 6) out[r] += dinv[r]*dinv[c] * X_hat[c]  — one wave per edge.
//    Lane f handles features [4f, 4f+3]: one B128 gather + 4 f32 atomics.
// ---------------------------------------------------------------------------
__global__ __launch_bounds__(256) void spmm_edge_kernel(const int* __restrict__ rows,
                                                        const int* __restrict__ cols,
                                                        const float* __restrict__ dinv,
                                                        const float* __restrict__ Xhat,
                                                        float* __restrict__ out, int E) {
    int e = blockIdx.x * 8 + (threadIdx.x >> 5);
    if (e >= E) return;
    int lane = threadIdx.x & 31;
    int r = rows[e];
    int c = cols[e];
    float val = dinv[r] * dinv[c];

    const float4 x = *(const float4*)(Xhat + (size_t)c * 128 + lane * 4);
    float* o = out + (size_t)r * 128 + lane * 4;
    atomicAdd(o + 0, val * x.x);
    atomicAdd(o + 1, val * x.y);
    atomicAdd(o + 2, val * x.z);
    atomicAdd(o + 3, val * x.w);
}

// ---------------------------------------------------------------------------
extern "C" void kernel_launch(void* const* d_in, const int* in_sizes, int n_in,
                              void* d_out, int out_size, void* d_ws, size_t ws_size,
                              hipStream_t stream) {
    const float* X  = (const float*)d_in[0];   // [N,128] fp32
    const float* W  = (const float*)d_in[1];   // [128,128] fp32
    const int*   ei = (const int*)d_in[2];     // [2,E] int32
    float* out = (float*)d_out;                // [N,128] fp32

    const int N = in_sizes[0] / 128;
    const int E = in_sizes[2] / 2;
    const int* rows = ei;
    const int* cols = ei + E;

    // workspace layout: X_hat [N*128] | deg [N] | dinv [N]
    float* Xhat = (float*)d_ws;
    float* deg  = Xhat + (size_t)N * 128;
    float* dinv = deg + N;

    deg_init_kernel<<<(N + 255) / 256, 256, 0, stream>>>(deg, N);
    deg_count_kernel<<<(E + 255) / 256, 256, 0, stream>>>(cols, deg, E);
    dinv_kernel<<<(N + 255) / 256, 256, 0, stream>>>(deg, dinv, N);

    gemm_xwT_wmma_kernel<<<(N + 15) / 16, 256, 0, stream>>>(X, W, Xhat, N);

    const int total = N * 128;
    out_init_kernel<<<(total + 255) / 256, 256, 0, stream>>>(dinv, Xhat, out, total);

    spmm_edge_kernel<<<(E + 7) / 8, 256, 0, stream>>>(rows, cols, dinv, Xhat, out, E);
}